// QModel_65481071409481
// MI455X (gfx1250) — compile-verified
//
#include <hip/hip_runtime.h>
#include <math.h>

#define N_WIRES 14
#define LAYERS  3
#define DIM     16384          // 2^14
#define TPB     512            // 16 wave32s
#define APT     (DIM / TPB)    // 32 amplitudes per thread
#define NWAVES  (TPB / 32)

typedef __attribute__((ext_vector_type(2))) float v2f;
typedef __attribute__((ext_vector_type(8))) float v8f;

struct c32 { float re, im; };

__device__ __forceinline__ c32 cmul(c32 a, c32 b) {
    c32 r; r.re = a.re * b.re - a.im * b.im; r.im = a.re * b.im + a.im * b.re; return r;
}

// Fused single-wire gate M = RZ(pz/2) * RX(px/2) (reference applies RX then RZ).
__device__ __forceinline__ void build_m(const float* __restrict__ params,
                                        int l, int w, c32 M[2][2]) {
    float px = params[(l * N_WIRES + w) * 2 + 0] * 0.5f;
    float pz = params[(l * N_WIRES + w) * 2 + 1] * 0.5f;
    float sx, cx, sz, cz;
    sincosf(px, &sx, &cx);
    sincosf(pz, &sz, &cz);
    M[0][0].re =  cx * cz;  M[0][0].im = -cx * sz;   // e^{-i hz} * cx
    M[0][1].re = -sx * sz;  M[0][1].im = -sx * cz;   // e^{-i hz} * (-i sx)
    M[1][0].re =  sx * sz;  M[1][0].im = -sx * cz;   // e^{+i hz} * (-i sx)
    M[1][1].re =  cx * cz;  M[1][1].im =  cx * sz;   // e^{+i hz} * cx
}

__global__ __launch_bounds__(TPB)
void qsim_kernel(const float* __restrict__ state_real,
                 const float* __restrict__ state_imag,
                 const float* __restrict__ params,
                 const float* __restrict__ head_w,
                 const float* __restrict__ head_b,
                 float* __restrict__ out)
{
    __shared__ float2 st[DIM];              // 128 KB statevector (re, im)
    __shared__ float  red[NWAVES * N_WIRES];

    const int  tid  = threadIdx.x;
    const int  b    = blockIdx.x;
    const int  lane = tid & 31;
    const int  wave = tid >> 5;
    const int  n    = lane & 15;            // WMMA column / A-row id
    const bool hi   = lane >= 16;

    // ---- encoding angle from state[:, -1] (only HBM traffic that matters) ----
    float ar = state_real[b * DIM + (DIM - 1)];
    float ai = state_imag[b * DIM + (DIM - 1)];
    float se, ce;
    sincosf(0.5f * atan2f(ai, ar), &se, &ce);

    // ---- RY encoding layer on |0..0>, same angle on all wires: product state ----
    for (int i = 0; i < APT; ++i) {
        int idx = tid + i * TPB;
        float amp = 1.0f;
        #pragma unroll
        for (int j = 0; j < N_WIRES; ++j)
            amp *= ((idx >> j) & 1) ? se : ce;
        st[idx] = make_float2(amp, 0.0f);
    }

    for (int l = 0; l < LAYERS; ++l) {
        // ---- 7 fused two-wire (RZ*RX) gate passes via V_WMMA_F32_16X16X4_F32 ----
        for (int g = 0; g < 7; ++g) {
            __syncthreads();
            c32 Ma[2][2], Mb[2][2];
            build_m(params, l, 2 * g + 0, Ma);   // more-significant quad bit
            build_m(params, l, 2 * g + 1, Mb);   // less-significant quad bit

            // B = G8^T (8x8 real gate, transposed) in columns n<8; rows split:
            // B1 holds K rows {kb, kb+1}, B2 rows {kb+4, kb+5}, kb = hi ? 2 : 0.
            // G8[2r+ri][2c+ci] from z = M2[r][c] = Ma[r>>1][c>>1]*Mb[r&1][c&1]:
            //   ri=0: (ci? -z.im : z.re)   ri=1: (ci? z.re : z.im)
            v2f B1, B2;
            {
                int kb = hi ? 2 : 0;
                int r  = n >> 1;
                int c1 = kb >> 1;                       // 0 or 1
                c32 zb = Mb[r & 1][c1];
                c32 z1 = cmul(Ma[r >> 1][0], zb);       // M2[r][c1]
                c32 z2 = cmul(Ma[r >> 1][1], zb);       // M2[r][c1+2]
                bool nodd = (n & 1) != 0;
                if (n < 8) {
                    B1.x = nodd ? z1.im : z1.re;        // k even
                    B1.y = nodd ? z1.re : -z1.im;       // k odd
                    B2.x = nodd ? z2.im : z2.re;
                    B2.y = nodd ? z2.re : -z2.im;
                } else {
                    B1.x = 0.f; B1.y = 0.f; B2.x = 0.f; B2.y = 0.f;
                }
            }

            const int pb = 12 - 2 * g;                  // bit position of wire 2g+1
            const int Sb = 1 << pb;                     // quad amps at i + a*Sb, a=0..3
            float* stf = reinterpret_cast<float*>(st);

            for (int t = 0; t < 16; ++t) {              // 4096 quads / (16 waves * 16)
                int qbase = wave * 256 + t * 16;
                int q  = qbase + n;
                int iq = ((q >> pb) << (pb + 2)) | (q & (Sb - 1));
                // A layout: VGPR0 = K0(lanes0-15)/K2(lanes16-31), VGPR1 = K1/K3
                float2 a1v = st[iq + (hi ? Sb : 0)];           // amp0 / amp1
                float2 a2v = st[iq + 2 * Sb + (hi ? Sb : 0)];  // amp2 / amp3
                v2f A1; A1.x = a1v.x; A1.y = a1v.y;
                v2f A2; A2.x = a2v.x; A2.y = a2v.y;
                v8f c0 = {};
                v8f d = __builtin_amdgcn_wmma_f32_16x16x4_f32(
                            false, A1, false, B1, (short)0, c0, false, false);
                d     = __builtin_amdgcn_wmma_f32_16x16x4_f32(
                            false, A2, false, B2, (short)0, d,  false, false);
                // D: VGPR rr, lanes0-15 -> (M=rr, N=lane); lanes16-31 -> (M=rr+8)
                if (n < 8) {
                    #pragma unroll
                    for (int rr = 0; rr < 8; ++rr) {
                        int m  = rr + (hi ? 8 : 0);
                        int qq = qbase + m;
                        int ii = ((qq >> pb) << (pb + 2)) | (qq & (Sb - 1));
                        int addr = ii + (n >> 1) * Sb;         // amp (n>>1) of quad
                        stf[addr * 2 + (n & 1)] = d[rr];       // .x / .y component
                    }
                }
            }
        }

        // ---- full circular-CNOT ring as one fused GF(2) permutation ----
        // src(b) = (b ^ (b>>1)) ^ ((b&1) ? 0x3000 : 0)
        __syncthreads();
        float2 tmp[APT];
        #pragma unroll
        for (int i = 0; i < APT; ++i) {
            int idx = tid + i * TPB;
            int src = (idx ^ (idx >> 1)) ^ ((idx & 1) ? 0x3000 : 0);
            tmp[i] = st[src];
        }
        __syncthreads();
        #pragma unroll
        for (int i = 0; i < APT; ++i)
            st[tid + i * TPB] = tmp[i];
    }

    // ---- PauliZ expectations: wire w <-> bit (13-w) ----
    __syncthreads();
    float acc[N_WIRES];
    #pragma unroll
    for (int w = 0; w < N_WIRES; ++w) acc[w] = 0.0f;
    for (int i = 0; i < APT; ++i) {
        int idx = tid + i * TPB;
        float2 a = st[idx];
        float p = a.x * a.x + a.y * a.y;
        #pragma unroll
        for (int w = 0; w < N_WIRES; ++w)
            acc[w] += (idx & (1 << (13 - w))) ? -p : p;
    }
    #pragma unroll
    for (int w = 0; w < N_WIRES; ++w) {
        float v = acc[w];
        for (int off = 16; off > 0; off >>= 1)
            v += __shfl_down(v, off, 32);
        acc[w] = v;
    }
    if (lane == 0) {
        #pragma unroll
        for (int w = 0; w < N_WIRES; ++w)
            red[wave * N_WIRES + w] = acc[w];
    }
    __syncthreads();
    if (tid == 0) {
        float r = head_b[0];
        #pragma unroll
        for (int w = 0; w < N_WIRES; ++w) {
            float e = 0.0f;
            for (int k = 0; k < NWAVES; ++k)
                e += red[k * N_WIRES + w];
            r += e * head_w[w];
        }
        out[b] = r;
    }
}

extern "C" void kernel_launch(void* const* d_in, const int* in_sizes, int n_in,
                              void* d_out, int out_size, void* d_ws, size_t ws_size,
                              hipStream_t stream) {
    const float* state_real = (const float*)d_in[0];
    const float* state_imag = (const float*)d_in[1];
    const float* params     = (const float*)d_in[2];
    const float* head_w     = (const float*)d_in[3];
    const float* head_b     = (const float*)d_in[4];
    float* out = (float*)d_out;
    (void)in_sizes; (void)n_in; (void)out_size; (void)d_ws; (void)ws_size;
    qsim_kernel<<<256, TPB, 0, stream>>>(state_real, state_imag, params,
                                         head_w, head_b, out);
}